// MultiTaskLoss_65042984731038
// MI455X (gfx1250) — compile-verified
//
#include <hip/hip_runtime.h>
#include <hip/hip_bf16.h>
#include <math.h>

// ---------------------------------------------------------------------------
// CDNA5 wave32 helpers
// ---------------------------------------------------------------------------
typedef float v2f __attribute__((ext_vector_type(2)));
typedef float v8f __attribute__((ext_vector_type(8)));

// Full-precision 32-lane sum using V_WMMA_F32_16X16X4_F32.
// A (16x4 f32, 2 VGPRs): lane m<16 holds A[m][0],A[m][1]; lane m+16 holds
// A[m][2],A[m][3]. Writing p to both VGPRs makes row m = {p_m,p_m,p_m16,p_m16}.
// With B = 0.5*ones(4x16):  D[m][n] = p_m + p_{m+16}  (exact f32 FMA chain).
// D VGPR j: lanes 0-15 = row j, lanes 16-31 = row j+8. Summing d[0..7] per
// lane then xor-16 gives the full wave sum in every lane.
__device__ __forceinline__ float wave_reduce32(float p) {
#if __has_builtin(__builtin_amdgcn_wmma_f32_16x16x4_f32)
  v2f a; a[0] = p;    a[1] = p;
  v2f b; b[0] = 0.5f; b[1] = 0.5f;
  v8f c = {0.f, 0.f, 0.f, 0.f, 0.f, 0.f, 0.f, 0.f};
  v8f d = __builtin_amdgcn_wmma_f32_16x16x4_f32(
      /*neg_a=*/false, a, /*neg_b=*/false, b,
      /*c_mod=*/(short)0, c, /*reuse_a=*/false, /*reuse_b=*/false);
  float h = ((d[0] + d[1]) + (d[2] + d[3])) + ((d[4] + d[5]) + (d[6] + d[7]));
  h += __shfl_xor(h, 16, 32);
  return h;
#else
  for (int off = 16; off > 0; off >>= 1) p += __shfl_xor(p, off, 32);
  return p;
#endif
}

__device__ __forceinline__ float bce_logit(float x, float t) {
  // max(x,0) - x*t + log1p(exp(-|x|)); hardware v_exp/v_log
  return fmaxf(x, 0.f) - x * t + __logf(1.f + __expf(-fabsf(x)));
}

// focal term, alpha=0.25, gamma=2 (pow collapses to a square)
__device__ __forceinline__ float focal_term(float x, bool is1) {
  float t  = is1 ? 1.f : 0.f;
  float p  = 1.f / (1.f + __expf(-x));
  float ce = bce_logit(x, t);
  float pt = is1 ? p : (1.f - p);
  float at = is1 ? 0.25f : 0.75f;
  float om = 1.f - pt;
  return at * om * om * ce;
}

// ---------------------------------------------------------------------------
// Kernel 1: segmentation BCE partial sums.  32*512*512 positions, 4 channels.
// 8192 blocks x 256 threads, 4 positions/thread (exact cover, no bounds).
// Per thread: 1 int4 mask load + 4 float4 logit loads (80 B, all b128).
// ---------------------------------------------------------------------------
__global__ void mtl_seg_kernel(const float* __restrict__ logits,
                               const int* __restrict__ mask,
                               float* __restrict__ partial) {
  const int tid = threadIdx.x;
  const size_t pos0 = ((size_t)blockIdx.x * 256 + tid) * 4;  // < 2^23
  const int    b    = (int)(pos0 >> 18);                     // / (512*512)
  const size_t rem  = pos0 & 262143;

  const int4 m4 = *(const int4*)(mask + (size_t)b * 262144 + rem);
  const float t0 = (float)m4.x, t1 = (float)m4.y, t2 = (float)m4.z, t3 = (float)m4.w;

  const float* base = logits + (size_t)b * 4 * 262144 + rem;
  float s = 0.f;
#pragma unroll
  for (int c = 0; c < 4; ++c) {
    float4 x = *(const float4*)(base + (size_t)c * 262144);
    s += bce_logit(x.x, t0) + bce_logit(x.y, t1) +
         bce_logit(x.z, t2) + bce_logit(x.w, t3);
  }

  float w = wave_reduce32(s);
  __shared__ float red[8];
  if ((tid & 31) == 0) red[tid >> 5] = w;
  __syncthreads();
  if (tid == 0) {
    float a = 0.f;
#pragma unroll
    for (int i = 0; i < 8; ++i) a += red[i];
    partial[blockIdx.x] = a;   // fixed-order => deterministic
  }
}

// ---------------------------------------------------------------------------
// Kernel 2: FCOS-style detection losses. One block per (scale, image).
// acc[blk*4 + {0,1,2,3}] = {cls_sum, reg_sum, cen_sum, npos}
// ---------------------------------------------------------------------------
__global__ void mtl_det_kernel(const float* __restrict__ cls0, const float* __restrict__ cls1,
                               const float* __restrict__ cls2,
                               const float* __restrict__ reg0, const float* __restrict__ reg1,
                               const float* __restrict__ reg2,
                               const float* __restrict__ cen0, const float* __restrict__ cen1,
                               const float* __restrict__ cen2,
                               const float* __restrict__ boxes, const int* __restrict__ labels,
                               float* __restrict__ acc) {
  const int blk = blockIdx.x;           // 0..95
  const int s   = blk >> 5;             // scale
  const int b   = blk & 31;             // image
  int H, logW, stridei;
  const float *cls, *reg, *cen;
  if (s == 0)      { H = 64; logW = 6; stridei = 8;  cls = cls0; reg = reg0; cen = cen0; }
  else if (s == 1) { H = 32; logW = 5; stridei = 16; cls = cls1; reg = reg1; cen = cen1; }
  else             { H = 16; logW = 4; stridei = 32; cls = cls2; reg = reg2; cen = cen2; }
  const int   W  = H;
  const int   L  = H * W;               // 4096 / 1024 / 256 (all %256 == 0)
  const float fs = (float)stridei;

  __shared__ float bx0[20], by0[20], bx1[20], by1[20], barea[20];
  __shared__ int   blab[20];
  const int tid = threadIdx.x;
  if (tid < 20) {
    const float* bp = boxes + ((size_t)b * 20 + tid) * 4;
    float cx = bp[0] * 512.f, cy = bp[1] * 512.f;
    float bw = bp[2] * 512.f, bh = bp[3] * 512.f;
    float x0 = cx - bw * 0.5f, y0 = cy - bh * 0.5f;
    float x1 = cx + bw * 0.5f, y1 = cy + bh * 0.5f;
    bx0[tid] = x0; by0[tid] = y0; bx1[tid] = x1; by1[tid] = y1;
    barea[tid] = (x1 - x0) * (y1 - y0);
    blab[tid]  = labels[b * 20 + tid];
  }
  __syncthreads();

  const float* clsb = cls + (size_t)b * 5 * L;
  const float* regb = reg + (size_t)b * 4 * L;
  const float* cenb = cen + (size_t)b * L;

  float cls_sum = 0.f, reg_sum = 0.f, cen_sum = 0.f, npos = 0.f;
  const float INF = __builtin_huge_valf();

  for (int l = tid; l < L; l += 256) {            // uniform trip count
    const int   h  = l >> logW, w = l & (W - 1);
    const float lx = ((float)w + 0.5f) * fs;
    const float ly = ((float)h + 0.5f) * fs;

    float best = INF;
    int   bi   = -1;
#pragma unroll 4
    for (int k = 0; k < 20; ++k) {                // first-min => jnp.argmin tie rule
      float lft = lx - bx0[k], top = ly - by0[k];
      float rgt = bx1[k] - lx, bot = by1[k] - ly;
      float mn  = fminf(fminf(lft, top), fminf(rgt, bot));
      float a   = (mn > 0.f) ? barea[k] : INF;
      if (a < best) { best = a; bi = k; }
    }
    const bool pos = (bi >= 0);
    const int  lab = pos ? blab[bi] : -1;

    float csum = 0.f;
#pragma unroll
    for (int j = 0; j < 4; ++j)                   // classes 1..4 of 5 channels
      csum += focal_term(clsb[(size_t)(j + 1) * L + l], j == lab);
    cls_sum += csum;

    if (pos) {
      npos += 1.f;
      const float lft = (lx - bx0[bi]) / fs, top = (ly - by0[bi]) / fs;
      const float rgt = (bx1[bi] - lx) / fs, bot = (by1[bi] - ly) / fs;
      // smooth-L1 (mean over 4 dims)
      float tgt[4] = {lft, top, rgt, bot};
      float sl = 0.f;
#pragma unroll
      for (int k = 0; k < 4; ++k) {
        float d  = regb[(size_t)k * L + l] - tgt[k];
        float ad = fabsf(d);
        sl += (ad < 1.f) ? 0.5f * d * d : ad - 0.5f;
      }
      reg_sum += sl * 0.25f;
      // centerness target + BCE
      const float eps  = 1e-6f;
      float r = (fminf(lft, rgt) / (fmaxf(lft, rgt) + eps)) *
                (fminf(top, bot) / (fmaxf(top, bot) + eps));
      r = fminf(fmaxf(r, 0.f), 1.f);
      cen_sum += bce_logit(cenb[l], sqrtf(r));
    }
  }

  // WMMA wave reduce each accumulator, then fixed-order cross-wave combine.
  float w0 = wave_reduce32(cls_sum);
  float w1 = wave_reduce32(reg_sum);
  float w2 = wave_reduce32(cen_sum);
  float w3 = wave_reduce32(npos);
  __shared__ float red[8][4];
  if ((tid & 31) == 0) {
    int wid = tid >> 5;
    red[wid][0] = w0; red[wid][1] = w1; red[wid][2] = w2; red[wid][3] = w3;
  }
  __syncthreads();
  if (tid == 0) {
    float a0 = 0.f, a1 = 0.f, a2 = 0.f, a3 = 0.f;
#pragma unroll
    for (int i = 0; i < 8; ++i) { a0 += red[i][0]; a1 += red[i][1]; a2 += red[i][2]; a3 += red[i][3]; }
    float* o = acc + (size_t)blk * 4;
    o[0] = a0; o[1] = a1; o[2] = a2; o[3] = a3;
  }
}

// ---------------------------------------------------------------------------
// Kernel 3: finalize — seg partial sum, cls focal (32x10), det combine, scalar.
// ---------------------------------------------------------------------------
__global__ void mtl_final_kernel(const float* __restrict__ ws,
                                 const float* __restrict__ cls_pred,
                                 const int* __restrict__ cls_target,
                                 float* __restrict__ out) {
  const int tid = threadIdx.x;

  float s = 0.f;
  for (int i = tid; i < 8192; i += 256) s += ws[i];
  float segw = wave_reduce32(s);

  float f = 0.f;
  for (int i = tid; i < 320; i += 256) {
    int bb = i / 10, j = i % 10;
    f += focal_term(cls_pred[bb * 10 + j], j == cls_target[bb]);
  }
  float clsw = wave_reduce32(f);

  __shared__ float redA[8], redB[8];
  if ((tid & 31) == 0) { redA[tid >> 5] = segw; redB[tid >> 5] = clsw; }
  __syncthreads();

  if (tid == 0) {
    float seg_sum = 0.f, cls_sum = 0.f;
#pragma unroll
    for (int i = 0; i < 8; ++i) { seg_sum += redA[i]; cls_sum += redB[i]; }

    const float seg_loss = seg_sum / (32.f * 4.f * 512.f * 512.f);
    const float cls_loss = cls_sum / 320.f;

    const float* acc = ws + 8192;
    float tot_cls = 0.f, tot_reg = 0.f, tot_cen = 0.f;
    for (int ss = 0; ss < 3; ++ss) {
      const float Lc = (ss == 0) ? 4096.f * 4.f : (ss == 1 ? 1024.f * 4.f : 256.f * 4.f);
      float cl = 0.f, rl = 0.f, cel = 0.f;
      for (int b = 0; b < 32; ++b) {
        const float* a = acc + (size_t)(ss * 32 + b) * 4;
        cl += a[0] / Lc;
        float np = a[3];
        if (np > 0.f) { rl += a[1] / np; cel += a[2] / np; }
      }
      tot_cls += cl / 32.f; tot_reg += rl / 32.f; tot_cen += cel / 32.f;
    }
    const float det_loss = (tot_cls + tot_reg + tot_cen) / 3.f;

    out[0] = 1.0f * seg_loss + 1.0f * det_loss + 0.5f * cls_loss;
  }
}

// ---------------------------------------------------------------------------
extern "C" void kernel_launch(void* const* d_in, const int* in_sizes, int n_in,
                              void* d_out, int out_size, void* d_ws, size_t ws_size,
                              hipStream_t stream) {
  (void)in_sizes; (void)n_in; (void)out_size; (void)ws_size;
  const float* seg_logits = (const float*)d_in[0];
  const int*   seg_mask   = (const int*)d_in[1];
  const float* cls0       = (const float*)d_in[2];
  const float* cls1       = (const float*)d_in[3];
  const float* cls2       = (const float*)d_in[4];
  const float* reg0       = (const float*)d_in[5];
  const float* reg1       = (const float*)d_in[6];
  const float* reg2       = (const float*)d_in[7];
  const float* cen0       = (const float*)d_in[8];
  const float* cen1       = (const float*)d_in[9];
  const float* cen2       = (const float*)d_in[10];
  const float* boxes      = (const float*)d_in[11];
  const int*   labels     = (const int*)d_in[12];
  const float* cls_pred   = (const float*)d_in[13];
  const int*   cls_target = (const int*)d_in[14];

  float* ws  = (float*)d_ws;    // [0,8192): seg partials; [8192,8576): det acc
  float* out = (float*)d_out;

  mtl_seg_kernel<<<8192, 256, 0, stream>>>(seg_logits, seg_mask, ws);
  mtl_det_kernel<<<96, 256, 0, stream>>>(cls0, cls1, cls2, reg0, reg1, reg2,
                                         cen0, cen1, cen2, boxes, labels, ws + 8192);
  mtl_final_kernel<<<1, 256, 0, stream>>>(ws, cls_pred, cls_target, out);
}